// LearningModel_22720376996168
// MI455X (gfx1250) — compile-verified
//
#include <hip/hip_runtime.h>
#include <math.h>

typedef __attribute__((ext_vector_type(2))) float v2f;
typedef __attribute__((ext_vector_type(8))) float v8f;

#define NROW 20000
#define DIN  128
#define HDIM 2048
#define EDIM 2048
#define SEV  2000
#define TPM_C 0.1f
#define ENT_C 0.01f

// workspace layout (float offsets)
#define WS_W2K    0
#define WS_B2K    2048
#define WS_LOGITS 2052
#define WS_EVGOOD (WS_LOGITS + NROW)
#define WS_EVHAS  (WS_EVGOOD + SEV)
#define WS_EVTIME (WS_EVHAS + SEV)
#define WS_EVENT  (WS_EVTIME + SEV)
#define WS_W1T    (WS_EVENT + SEV)          // 8-byte aligned; 262144 floats

// ---------- w2k = W2 @ key ----------
__global__ __launch_bounds__(256) void k_w2k(const float* __restrict__ W2,
                                             const float* __restrict__ key,
                                             float* __restrict__ w2k) {
    int h = blockIdx.x * 256 + threadIdx.x;
    if (h >= HDIM) return;
    const float4* row = (const float4*)(W2 + (size_t)h * EDIM);
    const float4* kv  = (const float4*)key;
    float acc = 0.f;
    for (int i = 0; i < EDIM / 4; ++i) {
        float4 a = row[i], b = kv[i];
        acc += a.x * b.x + a.y * b.y + a.z * b.z + a.w * b.w;
    }
    w2k[h] = acc;
}

// ---------- b2k = b2 . key ----------
__global__ __launch_bounds__(256) void k_b2k(const float* __restrict__ b2,
                                             const float* __restrict__ key,
                                             float* __restrict__ out) {
    __shared__ float red[8];
    float acc = 0.f;
    for (int i = threadIdx.x; i < EDIM; i += 256) acc += b2[i] * key[i];
    for (int o = 16; o; o >>= 1) acc += __shfl_xor(acc, o, 32);
    if ((threadIdx.x & 31) == 0) red[threadIdx.x >> 5] = acc;
    __syncthreads();
    if (threadIdx.x == 0) {
        float s = 0.f;
        for (int w = 0; w < 8; ++w) s += red[w];
        *out = s;
    }
}

// ---------- pre-swizzle W1 into WMMA B-fragment order ----------
// W1t[chunk][ks][lane] = float2{ W1[kk][col], W1[kk+1][col] }
//   kk  = ks*4 + (lane>=16 ? 2 : 0),  col = chunk*16 + (lane&15)
// -> per k-step B fragment is one coalesced 256B global_load_b64 record.
__global__ __launch_bounds__(256) void k_w1t(const float* __restrict__ W1,
                                             float2* __restrict__ W1t) {
    int idx = blockIdx.x * 256 + threadIdx.x;       // 128*32*32 = 131072
    if (idx >= (HDIM / 16) * (DIN / 4) * 32) return;
    int lane  = idx & 31;
    int ks    = (idx >> 5) & 31;
    int chunk = idx >> 10;
    int kk  = ks * 4 + (lane >> 4) * 2;
    int col = chunk * 16 + (lane & 15);
    float2 v;
    v.x = W1[(size_t)kk * HDIM + col];
    v.y = W1[(size_t)(kk + 1) * HDIM + col];
    W1t[idx] = v;
}

// ---------- logits[n] = relu(x_n@W1 + b1) . w2k + b2k  (WMMA f32) ----------
__global__ __launch_bounds__(128) void k_logits(const float* __restrict__ X,
                                                const float2* __restrict__ W1t,
                                                const float* __restrict__ b1,
                                                const float* __restrict__ w2k,
                                                const float* __restrict__ b2k,
                                                float* __restrict__ logits) {
    __shared__ float At[16 * DIN];       // 16-row x 128-col X tile (8 KB)
    __shared__ float red[4][16];         // per-wave row partials
    const int tid  = threadIdx.x;
    const int wave = tid >> 5;
    const int lane = tid & 31;
    const int row0 = blockIdx.x * 16;

    // stage X tile into LDS (contiguous 2048 floats)
    {
        const float4* src = (const float4*)(X + (size_t)row0 * DIN);
        float4* dst = (float4*)At;
        for (int i = tid; i < (16 * DIN) / 4; i += 128) dst[i] = src[i];
    }
    __syncthreads();

    const int hh  = lane >> 4;           // half: 0 = lanes 0-15, 1 = lanes 16-31
    const int l16 = lane & 15;

    // Hoist all A fragments (chunk-invariant): 32 k-steps x v2f = 64 VGPRs.
    // A 16x4 f32 layout: lanes 0-15 -> M=lane, K=kk+v ; lanes 16-31 -> K=kk+2+v
    v2f afrag[32];
#pragma unroll
    for (int ks = 0; ks < 32; ++ks) {
        const int kk = ks * 4 + hh * 2;
        afrag[ks].x = At[l16 * DIN + kk];
        afrag[ks].y = At[l16 * DIN + kk + 1];
    }

    float racc[8];
#pragma unroll
    for (int r = 0; r < 8; ++r) racc[r] = 0.f;

    // each wave handles H-chunks wave, wave+4, ... (128 chunks of 16 cols)
    for (int chunk = wave; chunk < HDIM / 16; chunk += 4) {
        const int c16 = chunk * 16;
        const int col = c16 + l16;
        const float2* Bp = W1t + ((size_t)chunk * 32) * 32 + lane;

        v8f c0, c1;                       // two independent WMMA chains
#pragma unroll
        for (int r = 0; r < 8; ++r) { c0[r] = 0.f; c1[r] = 0.f; }

#pragma unroll
        for (int ks = 0; ks < 32; ks += 2) {
            float2 bb0 = Bp[(ks)     * 32];
            float2 bb1 = Bp[(ks + 1) * 32];
            v2f b0, b1f;
            b0.x = bb0.x;  b0.y = bb0.y;
            b1f.x = bb1.x; b1f.y = bb1.y;
            c0 = __builtin_amdgcn_wmma_f32_16x16x4_f32(false, afrag[ks], false, b0,
                                                       (short)0, c0, false, false);
            c1 = __builtin_amdgcn_wmma_f32_16x16x4_f32(false, afrag[ks + 1], false, b1f,
                                                       (short)0, c1, false, false);
        }

        const float wv = w2k[col];
        const float bv = b1[col];
#pragma unroll
        for (int r = 0; r < 8; ++r) racc[r] += fmaxf(c0[r] + c1[r] + bv, 0.f) * wv;
    }

    // reduce each row partial across the 16 lanes of the half (columns)
#pragma unroll
    for (int r = 0; r < 8; ++r) {
        float p = racc[r];
        for (int off = 8; off; off >>= 1) p += __shfl_xor(p, off, 16);
        if (l16 == 0) red[wave][hh * 8 + r] = p;   // half0 -> M=r, half1 -> M=r+8
    }
    __syncthreads();
    if (tid < 16) {
        float s = red[0][tid] + red[1][tid] + red[2][tid] + red[3][tid];
        logits[row0 + tid] = s + b2k[0];
    }
}

// ---------- per-event masked log-softmax statistics ----------
__device__ inline void lse_merge(float& m, float& Z, float& T,
                                 float m2, float Z2, float T2) {
    float M = fmaxf(m, m2);
    if (M == -INFINITY) { m = -INFINITY; Z = 0.f; T = 0.f; return; }
    float s1 = __expf(m - M);
    float s2 = __expf(m2 - M);
    Z = Z * s1 + Z2 * s2;
    T = T * s1 + T2 * s2;
    m = M;
}

__device__ inline void online_acc(float l, float& m, float& Z, float& T) {
    if (l > m) {                       // online softmax rescale
        float sc = __expf(m - l);      // exp(-inf)=0 on first hit
        Z = Z * sc + 1.f;
        T = T * sc + l;
        m = l;
    } else {
        float e = __expf(l - m);
        Z += e;
        T += e * l;
    }
}

__global__ __launch_bounds__(256) void k_events(const unsigned char* __restrict__ mask,
                                                const unsigned char* __restrict__ good,
                                                const int* __restrict__ sel_idx,
                                                const float* __restrict__ times,
                                                const float* __restrict__ logits,
                                                float* __restrict__ ev_good,
                                                float* __restrict__ ev_has,
                                                float* __restrict__ ev_time,
                                                float* __restrict__ ev_ent) {
    const int s = blockIdx.x;
    const uchar4* mrow4 = (const uchar4*)(mask + (size_t)s * NROW);
    const uchar4* good4 = (const uchar4*)good;
    const float4* lg4   = (const float4*)logits;

    float m = -INFINITY, Z = 0.f, T = 0.f, sg = 0.f;
    int ng = 0, psz = 0;

    for (int q = threadIdx.x; q < NROW / 4; q += 256) {
        uchar4 mq = mrow4[q];
        if (mq.x | mq.y | mq.z | mq.w) {
            float4 lq = lg4[q];
            uchar4 gq = good4[q];
            if (mq.x) { ++psz; online_acc(lq.x, m, Z, T); if (gq.x) { ++ng; sg += lq.x; } }
            if (mq.y) { ++psz; online_acc(lq.y, m, Z, T); if (gq.y) { ++ng; sg += lq.y; } }
            if (mq.z) { ++psz; online_acc(lq.z, m, Z, T); if (gq.z) { ++ng; sg += lq.z; } }
            if (mq.w) { ++psz; online_acc(lq.w, m, Z, T); if (gq.w) { ++ng; sg += lq.w; } }
        }
    }

    // warp butterfly reduce
    for (int off = 16; off; off >>= 1) {
        float om = __shfl_xor(m, off, 32);
        float oZ = __shfl_xor(Z, off, 32);
        float oT = __shfl_xor(T, off, 32);
        sg  += __shfl_xor(sg, off, 32);
        ng  += __shfl_xor(ng, off, 32);
        psz += __shfl_xor(psz, off, 32);
        lse_merge(m, Z, T, om, oZ, oT);
    }

    __shared__ float sm[8], sZ[8], sT[8], ssg[8];
    __shared__ int sng[8], spsz[8];
    if ((threadIdx.x & 31) == 0) {
        int w = threadIdx.x >> 5;
        sm[w] = m; sZ[w] = Z; sT[w] = T; ssg[w] = sg; sng[w] = ng; spsz[w] = psz;
    }
    __syncthreads();

    if (threadIdx.x == 0) {
        float M = sm[0], ZZ = sZ[0], TT = sT[0], SG = ssg[0];
        int NG = sng[0], PS = spsz[0];
        for (int w = 1; w < 8; ++w) {          // fixed-order merge: deterministic
            lse_merge(M, ZZ, TT, sm[w], sZ[w], sT[w]);
            SG += ssg[w]; NG += sng[w]; PS += spsz[w];
        }
        float L = M + __logf(ZZ);              // logsumexp over passive set
        bool hg = NG > 0;
        ev_good[s] = hg ? (NG * L - SG) / (float)NG : 0.f;
        ev_has[s]  = hg ? 1.f : 0.f;
        float sel_lsm = logits[sel_idx[s]] - L;
        ev_time[s] = times[s] * sel_lsm;
        ev_ent[s]  = (TT / ZZ - L) / __logf((float)PS);
    }
}

// ---------- final deterministic reduction over S ----------
__global__ __launch_bounds__(256) void k_final(const float* __restrict__ ev_good,
                                               const float* __restrict__ ev_has,
                                               const float* __restrict__ ev_time,
                                               const float* __restrict__ ev_ent,
                                               const float* __restrict__ times,
                                               float* __restrict__ out) {
    __shared__ float red[8][5];
    float g = 0.f, h = 0.f, t = 0.f, e = 0.f, tv = 0.f;
    for (int s = threadIdx.x; s < SEV; s += 256) {
        g += ev_good[s]; h += ev_has[s]; t += ev_time[s]; e += ev_ent[s]; tv += times[s];
    }
    for (int off = 16; off; off >>= 1) {
        g  += __shfl_xor(g, off, 32);
        h  += __shfl_xor(h, off, 32);
        t  += __shfl_xor(t, off, 32);
        e  += __shfl_xor(e, off, 32);
        tv += __shfl_xor(tv, off, 32);
    }
    if ((threadIdx.x & 31) == 0) {
        int w = threadIdx.x >> 5;
        red[w][0] = g; red[w][1] = h; red[w][2] = t; red[w][3] = e; red[w][4] = tv;
    }
    __syncthreads();
    if (threadIdx.x == 0) {
        float G = 0.f, H = 0.f, Tt = 0.f, E = 0.f, TV = 0.f;
        for (int w = 0; w < 8; ++w) {
            G += red[w][0]; H += red[w][1]; Tt += red[w][2]; E += red[w][3]; TV += red[w][4];
        }
        out[0] = G;                 // good_loss
        out[1] = H;                 // num_good_steps (as float)
        out[2] = TPM_C * Tt;        // time_loss
        out[3] = TV;                // time_volume
        out[4] = ENT_C * E;         // entropy_loss
        out[5] = (float)SEV;        // num_steps
    }
}

extern "C" void kernel_launch(void* const* d_in, const int* in_sizes, int n_in,
                              void* d_out, int out_size, void* d_ws, size_t ws_size,
                              hipStream_t stream) {
    const float*         X    = (const float*)d_in[0];          // [N, 128]
    const float*         W1   = (const float*)d_in[1];          // [128, 2048]
    const float*         b1   = (const float*)d_in[2];          // [2048]
    const float*         W2   = (const float*)d_in[3];          // [2048, 2048]
    const float*         b2   = (const float*)d_in[4];          // [2048]
    const float*         key  = (const float*)d_in[5];          // [2048]
    const float*         times= (const float*)d_in[6];          // [S]
    const unsigned char* mask = (const unsigned char*)d_in[7];  // [S, N] bool
    const unsigned char* good = (const unsigned char*)d_in[8];  // [N] bool
    const int*           sel  = (const int*)d_in[9];            // [S]
    float* out = (float*)d_out;
    float* ws  = (float*)d_ws;

    k_w2k   <<<HDIM / 256, 256, 0, stream>>>(W2, key, ws + WS_W2K);
    k_b2k   <<<1, 256, 0, stream>>>(b2, key, ws + WS_B2K);
    k_w1t   <<<(HDIM / 16) * (DIN / 4) * 32 / 256, 256, 0, stream>>>(
                W1, (float2*)(ws + WS_W1T));
    k_logits<<<NROW / 16, 128, 0, stream>>>(X, (const float2*)(ws + WS_W1T), b1,
                                            ws + WS_W2K, ws + WS_B2K, ws + WS_LOGITS);
    k_events<<<SEV, 256, 0, stream>>>(mask, good, sel, times, ws + WS_LOGITS,
                                      ws + WS_EVGOOD, ws + WS_EVHAS,
                                      ws + WS_EVTIME, ws + WS_EVENT);
    k_final <<<1, 256, 0, stream>>>(ws + WS_EVGOOD, ws + WS_EVHAS, ws + WS_EVTIME,
                                    ws + WS_EVENT, times, out);
}